// CapsuleBlock_33672543600996
// MI455X (gfx1250) — compile-verified
//
#include <hip/hip_runtime.h>

// Problem constants (from setup_inputs)
#define B_TOT   64
#define N_IN    2048
#define D_IN    8
#define N_OUT   16
#define D_OUT   16
#define BT      16                      // batch tile per block (one WMMA M)
#define NPB     32                      // input capsules per block
#define NCHUNK  (N_IN / NPB)            // 64
#define S_ELEMS (B_TOT * N_OUT * D_OUT) // 16384 floats per s buffer
#define WN_STRIDE (N_OUT * D_IN * D_OUT) // 2048 floats between consecutive n in W

typedef float v2f __attribute__((ext_vector_type(2)));
typedef float v8f __attribute__((ext_vector_type(8)));

// xor-swizzle add within a 32-lane group: and=0x1f, or=0, xor=MASK
template <int MASK>
__device__ __forceinline__ float xor_add(float v) {
    const int sw = __builtin_amdgcn_ds_swizzle(__float_as_int(v), (MASK << 10) | 0x1f);
    return v + __int_as_float(sw);
}

// One routing pass. Per (n, k-wave):
//   hats tile  H[16b x 16o] = X (16x8) x W_nk (8x16)        -> 2 WMMAs
//   V[d,b]     = W_nk (8x16) x dv_k^T (16x16)               -> 4 WMMAs (B hoisted)
//   bias[b]    = sum_d x[b,d] * V[d,b]                      -> 8 per-lane FMAs
//   softmax over k via double-buffered LDS tile (1 barrier/n), c-weighted accumulate.
// Block: 32 x 16 (one wave per output capsule k). Grid: (B/16, NCHUNK).
__global__ __launch_bounds__(512)
void routing_pass(const float* __restrict__ x,      // [B, N_in, D_in]
                  const float* __restrict__ W,      // [N_in, N_out, D_in, D_out]
                  const float* __restrict__ dotvec, // [B, N_out, D_out]
                  float* __restrict__ partial,      // [NCHUNK, B, N_out, D_out]
                  int use_bias)
{
    const int bx    = blockIdx.x;        // batch tile 0..3
    const int chunk = blockIdx.y;        // 0..NCHUNK-1
    const int k     = threadIdx.y;       // output capsule / wave id 0..15
    const int lane  = threadIdx.x;       // 0..31
    const int hi    = lane >> 4;         // lane half
    const int lo    = lane & 15;
    const int dsel  = hi * 2;            // K-pair selector per ISA fp32 A/B layout

    __shared__ __align__(64) float biasLds[2][BT][N_OUT]; // double-buffered [b][k]
    __shared__ __align__(64) float cLds[N_OUT][BT];       // [k][b] (wave-local columns)

    v8f sAcc = {};                       // c-weighted hats accumulator (C layout)

    const int bBase = bx * BT;
    const int nBase = chunk * NPB;

    // Loop-invariant B fragments for V = W_nk x dv^T.
    // B element (K=o, N=b): VGPR0 row = 4c+dsel, VGPR1 row = 4c+dsel+1.
    v2f dvb[4] = {};
    if (use_bias) {
        const float* dvk = dotvec + ((size_t)(bBase + lo) * N_OUT + k) * D_OUT;
        #pragma unroll
        for (int c = 0; c < 4; ++c) {
            dvb[c].x = dvk[4 * c + dsel + 0];
            dvb[c].y = dvk[4 * c + dsel + 1];
        }
    }

    // Streaming pointers (advance by fixed strides per n)
    const float* xp = x + ((size_t)(bBase + lo) * N_IN + nBase) * D_IN;
    const float* wp = W + ((size_t)nBase * N_OUT + k) * (D_IN * D_OUT);
    const int  aoff = (lo & 7) * D_OUT + dsel;  // clamped (always in-bounds) A row for V
    const bool arowValid = (lo < D_IN);

    for (int ni = 0; ni < NPB; ++ni) {
        // x row (8 floats) for b = bBase+lo; feeds both the A fragment and bias dot
        const float4 xA = reinterpret_cast<const float4*>(xp)[0];
        const float4 xB = reinterpret_cast<const float4*>(xp)[1];

        // A fragment (16x4 per WMMA): lanes 0-15 K={0,1}, lanes 16-31 K={2,3}
        v2f a0, a1;
        a0.x = hi ? xA.z : xA.x;  a0.y = hi ? xA.w : xA.y;   // K = 0..3
        a1.x = hi ? xB.z : xB.x;  a1.y = hi ? xB.w : xB.y;   // K = 4..7

        // B fragment for hats: W[n,k,d,o], (K=d, N=o=lo)
        v2f b0, b1;
        b0.x = wp[(0 + dsel) * D_OUT + lo];
        b0.y = wp[(1 + dsel) * D_OUT + lo];
        b1.x = wp[(4 + dsel) * D_OUT + lo];
        b1.y = wp[(5 + dsel) * D_OUT + lo];

        v8f h = {};
        h = __builtin_amdgcn_wmma_f32_16x16x4_f32(false, a0, false, b0,
                                                  (short)0, h, false, false);
        h = __builtin_amdgcn_wmma_f32_16x16x4_f32(false, a1, false, b1,
                                                  (short)0, h, false, false);

        if (use_bias) {
            // V[d,b] = sum_o W_nk[d,o] * dv[b,k,o] via 16x16x16 (4 chained WMMAs).
            // A rows 8..15: clamped-address unconditional b64 loads, zeroed by cndmask.
            const float2* wa2 = reinterpret_cast<const float2*>(wp + aoff); // 8B aligned
            const float2 w0 = wa2[0];   // o = 4*0 + dsel + {0,1}
            const float2 w1 = wa2[2];   // o = 4*1 + dsel + {0,1}
            const float2 w2 = wa2[4];   // o = 4*2 + dsel + {0,1}
            const float2 w3 = wa2[6];   // o = 4*3 + dsel + {0,1}
            v2f wa[4];
            wa[0].x = arowValid ? w0.x : 0.f;  wa[0].y = arowValid ? w0.y : 0.f;
            wa[1].x = arowValid ? w1.x : 0.f;  wa[1].y = arowValid ? w1.y : 0.f;
            wa[2].x = arowValid ? w2.x : 0.f;  wa[2].y = arowValid ? w2.y : 0.f;
            wa[3].x = arowValid ? w3.x : 0.f;  wa[3].y = arowValid ? w3.y : 0.f;

            v8f Vt = {};
            #pragma unroll
            for (int c = 0; c < 4; ++c)
                Vt = __builtin_amdgcn_wmma_f32_16x16x4_f32(false, wa[c], false, dvb[c],
                                                           (short)0, Vt, false, false);

            // bias[b=lo] = sum_d x[b,n,d] * V[d,b]  (valid in lanes 0-15)
            const float bias =
                xA.x * Vt[0] + xA.y * Vt[1] + xA.z * Vt[2] + xA.w * Vt[3] +
                xB.x * Vt[4] + xB.y * Vt[5] + xB.z * Vt[6] + xB.w * Vt[7];
            const int p = ni & 1;
            if (lane < 16) biasLds[p][lo][k] = bias;
            __syncthreads();   // single barrier per n (double-buffered biasLds)

            // softmax over k for batch row b = lo (b128 row reads)
            const float4 r0 = *reinterpret_cast<const float4*>(&biasLds[p][lo][0]);
            const float4 r1 = *reinterpret_cast<const float4*>(&biasLds[p][lo][4]);
            const float4 r2 = *reinterpret_cast<const float4*>(&biasLds[p][lo][8]);
            const float4 r3 = *reinterpret_cast<const float4*>(&biasLds[p][lo][12]);
            const float bk = biasLds[p][lo][k];
            float mx = fmaxf(fmaxf(fmaxf(r0.x, r0.y), fmaxf(r0.z, r0.w)),
                             fmaxf(fmaxf(r1.x, r1.y), fmaxf(r1.z, r1.w)));
            mx = fmaxf(mx, fmaxf(fmaxf(fmaxf(r2.x, r2.y), fmaxf(r2.z, r2.w)),
                                 fmaxf(fmaxf(r3.x, r3.y), fmaxf(r3.z, r3.w))));
            const float den =
                __expf(r0.x - mx) + __expf(r0.y - mx) + __expf(r0.z - mx) + __expf(r0.w - mx) +
                __expf(r1.x - mx) + __expf(r1.y - mx) + __expf(r1.z - mx) + __expf(r1.w - mx) +
                __expf(r2.x - mx) + __expf(r2.y - mx) + __expf(r2.z - mx) + __expf(r2.w - mx) +
                __expf(r3.x - mx) + __expf(r3.y - mx) + __expf(r3.z - mx) + __expf(r3.w - mx);
            const float cb = __expf(bk - mx) / den;   // c[b=lo][k]

            // wave-local c exchange: write column, read back as two b128s (in-order LDS)
            cLds[k][lo] = cb;
            const float4 c0 = *reinterpret_cast<const float4*>(&cLds[k][8 * hi + 0]);
            const float4 c1 = *reinterpret_cast<const float4*>(&cLds[k][8 * hi + 4]);
            sAcc[0] += c0.x * h[0];  sAcc[1] += c0.y * h[1];
            sAcc[2] += c0.z * h[2];  sAcc[3] += c0.w * h[3];
            sAcc[4] += c1.x * h[4];  sAcc[5] += c1.y * h[5];
            sAcc[6] += c1.z * h[6];  sAcc[7] += c1.w * h[7];
        } else {
            // iteration 0: bias == 0 -> c == 1/16 uniformly
            #pragma unroll
            for (int v = 0; v < 8; ++v) sAcc[v] += 0.0625f * h[v];
        }

        xp += D_IN;
        wp += WN_STRIDE;
    }

    // deterministic per-chunk partial write (no float atomics)
    float* pout = partial + (size_t)chunk * S_ELEMS;
    #pragma unroll
    for (int v = 0; v < 8; ++v) {
        const int b_ = bBase + v + 8 * hi;
        pout[((size_t)b_ * N_OUT + k) * D_OUT + lo] = sAcc[v];
    }
}

// Sum partials over chunks, squash per capsule (16-lane reduction over o).
// mode 0: out = squash(s)            mode 1: out = addend + squash(s)
// mode 2: out = squash(s) (final)
__global__ __launch_bounds__(256)
void reduce_squash(const float* __restrict__ partial,
                   const float* __restrict__ addend,
                   float* __restrict__ out,
                   int mode)
{
    const int idx = blockIdx.x * 256 + threadIdx.x;   // b*256 + k*16 + o
    float s = 0.f;
    #pragma unroll 8
    for (int ch = 0; ch < NCHUNK; ++ch)
        s += partial[(size_t)ch * S_ELEMS + idx];

    float s2 = s * s;                                 // sum over o (16 lanes; masks < 16
    s2 = xor_add<1>(s2);                              //  stay inside each o-group)
    s2 = xor_add<2>(s2);
    s2 = xor_add<4>(s2);
    s2 = xor_add<8>(s2);

    const float scale = (s2 / (1.0f + s2)) * __frsqrt_rn(s2);
    float o = scale * s;
    if (mode == 1) o += addend[idx];
    out[idx] = o;
}

extern "C" void kernel_launch(void* const* d_in, const int* in_sizes, int n_in,
                              void* d_out, int out_size, void* d_ws, size_t ws_size,
                              hipStream_t stream) {
    (void)in_sizes; (void)n_in; (void)out_size; (void)ws_size;
    const float* x = (const float*)d_in[0];
    const float* W = (const float*)d_in[1];
    float* out = (float*)d_out;

    float* partial = (float*)d_ws;                         // NCHUNK * 16384 floats (4 MB)
    float* out0    = partial + (size_t)NCHUNK * S_ELEMS;   // 64 KB
    float* outsum  = out0 + S_ELEMS;                       // 64 KB

    dim3 grid(B_TOT / BT, NCHUNK);   // (4, 64)
    dim3 block(32, 16);              // 16 waves, one per output capsule

    // iteration 0: c = 1/16
    routing_pass<<<grid, block, 0, stream>>>(x, W, outsum /*unused*/, partial, 0);
    reduce_squash<<<S_ELEMS / 256, 256, 0, stream>>>(partial, outsum, out0, 0);

    // iteration 1: bias1 = <hats, out0>
    routing_pass<<<grid, block, 0, stream>>>(x, W, out0, partial, 1);
    reduce_squash<<<S_ELEMS / 256, 256, 0, stream>>>(partial, out0, outsum, 1); // out0+out1

    // iteration 2: bias2 = <hats, out0 + out1>
    routing_pass<<<grid, block, 0, stream>>>(x, W, outsum, partial, 1);
    reduce_squash<<<S_ELEMS / 256, 256, 0, stream>>>(partial, outsum /*unused*/, out, 2);
}